// DiscreteMapSmoother_5987184411250
// MI455X (gfx1250) — compile-verified
//
#include <hip/hip_runtime.h>
#include <cstdint>
#include <cstddef>

// Depthwise 5x5 SAME conv, NHWC (16,512,512,8) f32 + bias.
// Memory-bound: ~268 MB moved -> ~11.5 us floor @ 23.3 TB/s.
// CDNA5 path: async global->LDS staging (ASYNCcnt) + swizzled LDS tile,
// packed f32 FMAs, non-temporal output stores (preserve input in L2 for
// halo reuse between neighboring tiles).

#define SW 512
#define SH 512
#define CH 8
#define TILE_W 32
#define TILE_H 32
#define HALO 2
#define IN_W (TILE_W + 2 * HALO) // 36
#define IN_H (TILE_H + 2 * HALO) // 36
#define NTHREADS 256

typedef __attribute__((ext_vector_type(4))) float v4f; // native vec for NT store

// Column swizzle: bijection on [0,36). Within a wave the strip index s is the
// varying term; f(2s+j) = (j&1)*18 + s + (j>>1), so 2*f+h covers all 16
// bank-quads exactly twice per ds_load_b128 -> conflict-free (2-cycle) LDS.
__device__ __forceinline__ int fswz(int col) {
  return (col & 1) * (IN_W / 2) + (col >> 1);
}

__global__ __launch_bounds__(NTHREADS)
void dwconv5x5_bias_kernel(const float* __restrict__ x,
                           const float* __restrict__ kern,   // (5,5,1,8)
                           const float* __restrict__ bias,   // (8,)
                           float* __restrict__ out) {
  __shared__ __align__(16) float lds_tile[IN_H * IN_W * CH]; // 41,472 B

  const int tid = threadIdx.x;
  const int tx = blockIdx.x;   // 16 tiles in x
  const int ty = blockIdx.y;   // 16 tiles in y
  const int b  = blockIdx.z;   // batch
  const int gx0 = tx * TILE_W - HALO;
  const int gy0 = ty * TILE_H - HALO;

  // Low 32 bits of the flat address of an LDS object == LDS byte offset
  // (aperture rule, ISA 10.2) -> usable as VDST address for async-to-LDS.
  const uint32_t lds_base = (uint32_t)(uintptr_t)(&lds_tile[0]);

  // ---- Stage input tile: global -> LDS via gfx1250 async copies ----------
  // 36*36 pixels * 2 halves = 2592 b128 chunks; ~10 per thread.
  for (int n = tid; n < IN_H * IN_W * 2; n += NTHREADS) {
    const int h2  = n & 1;        // which float4 half of the 8 channels
    const int pix = n >> 1;
    const int col = pix % IN_W;
    const int r   = pix / IN_W;
    const int gy = gy0 + r;
    const int gx = gx0 + col;
    const int ldx = (r * IN_W + fswz(col)) * CH + 4 * h2;
    if ((unsigned)gy < (unsigned)SH && (unsigned)gx < (unsigned)SW) {
      const float* gp =
          x + ((((size_t)b * SH + gy) * SW + gx) * CH + 4 * h2);
      const uint32_t loff = lds_base + (uint32_t)(ldx * 4);
      // Per-lane async copy, tracked by ASYNCcnt (EXEC-masked by the branch).
      asm volatile("global_load_async_to_lds_b128 %0, %1, off"
                   :: "v"(loff), "v"((unsigned long long)(uintptr_t)gp)
                   : "memory");
    } else {
      // SAME zero padding: OOB chunks are zero-filled via DS stores.
      *(float4*)&lds_tile[ldx] = make_float4(0.f, 0.f, 0.f, 0.f);
    }
  }
  // Each wave drains its own async queue; barrier publishes LDS to the WG.
  asm volatile("s_wait_asynccnt 0x0" ::: "memory");
  __syncthreads();

  // ---- Compute: thread = (4-ch half, 2-px strip, 4-row group) ------------
  const int h  = tid & 1;         // channel half: ch 4h..4h+3
  const int s  = (tid >> 1) & 15; // strip: output cols 2s..2s+1
  const int rg = tid >> 5;        // row group: output rows 4rg..4rg+3
  const int rbase = rg * 4;

  // 25 per-channel-quad weights in registers. kernel[(ky*5+kx)*8 + 4h .. +3]
  const float4* kv = (const float4*)kern;
  float4 wv[25];
#pragma unroll
  for (int i = 0; i < 25; ++i) wv[i] = kv[i * 2 + h];
  const float4 bv = ((const float4*)bias)[h];

  float4 acc[4][2];
#pragma unroll
  for (int oy = 0; oy < 4; ++oy)
#pragma unroll
    for (int px = 0; px < 2; ++px) acc[oy][px] = bv;

  // Slide over the 8 input rows this thread needs; each row feeds up to 4
  // output rows (oy = iy - ky, folded at compile time by full unrolling).
#pragma unroll
  for (int iy = 0; iy < 8; ++iy) {
    const int r = rbase + iy;
    float4 win[6];
#pragma unroll
    for (int j = 0; j < 6; ++j) {
      const int fc = (j & 1) * (IN_W / 2) + s + (j >> 1); // fswz(2s + j)
      win[j] = *(const float4*)&lds_tile[(r * IN_W + fc) * CH + 4 * h];
    }
#pragma unroll
    for (int ky = 0; ky < 5; ++ky) {
      const int oy = iy - ky;
      if (oy < 0 || oy > 3) continue;
#pragma unroll
      for (int kx = 0; kx < 5; ++kx) {
        const float4 wk = wv[ky * 5 + kx];
#pragma unroll
        for (int px = 0; px < 2; ++px) {
          const float4 v = win[px + kx];
          acc[oy][px].x = fmaf(wk.x, v.x, acc[oy][px].x);
          acc[oy][px].y = fmaf(wk.y, v.y, acc[oy][px].y);
          acc[oy][px].z = fmaf(wk.z, v.z, acc[oy][px].z);
          acc[oy][px].w = fmaf(wk.w, v.w, acc[oy][px].w);
        }
      }
    }
  }

  // ---- Store 4x2 output pixels (b128, non-temporal) ----------------------
  // Output is write-once/never-read: NT stores keep the input stream (which
  // neighbors re-read for halos) resident in the 192 MB L2.
#pragma unroll
  for (int oy = 0; oy < 4; ++oy) {
    const int gy = ty * TILE_H + rbase + oy;
#pragma unroll
    for (int px = 0; px < 2; ++px) {
      const int gx = tx * TILE_W + s * 2 + px;
      float* op = out + ((((size_t)b * SH + gy) * SW + gx) * CH + 4 * h);
      const v4f vv = { acc[oy][px].x, acc[oy][px].y,
                       acc[oy][px].z, acc[oy][px].w };
      __builtin_nontemporal_store(vv, (v4f*)op);
    }
  }
}

extern "C" void kernel_launch(void* const* d_in, const int* in_sizes, int n_in,
                              void* d_out, int out_size, void* d_ws, size_t ws_size,
                              hipStream_t stream) {
  const float* x    = (const float*)d_in[0];
  const float* kern = (const float*)d_in[1];
  const float* bias = (const float*)d_in[2];
  float* out = (float*)d_out;
  const int B = in_sizes[0] / (SH * SW * CH); // 16
  dim3 grid(SW / TILE_W, SH / TILE_H, B);     // (16,16,16)
  dim3 block(NTHREADS);
  dwconv5x5_bias_kernel<<<grid, block, 0, stream>>>(x, kern, bias, out);
}